// AD_MAE_6442450944038
// MI455X (gfx1250) — compile-verified
//
#include <hip/hip_runtime.h>

#define BINS 10
#define BLOCK 256
// float2 records per thread: 10 bins + 1 pad => 88B/thread stride.
// Bank analysis (wave32, 64 banks, b64 access): lane l, bin b touches bank pair
// {(22*l + 2*b) mod 64, +1}; 22*l mod 64 is distinct & even for l in [0,32) -> conflict-free.
#define STRIDE_REC 11

typedef __attribute__((ext_vector_type(4))) float v4f;

__global__ __launch_bounds__(BLOCK) void ghm_init_kernel(float* __restrict__ gsum,
                                                         unsigned int* __restrict__ gcnt,
                                                         float* __restrict__ out) {
  int t = threadIdx.x;
  if (t < BINS) { gsum[t] = 0.0f; gcnt[t] = 0u; }
  if (t == 0) out[0] = 0.0f;
}

__global__ __launch_bounds__(BLOCK) void ghm_hist_kernel(const float* __restrict__ pred,
                                                         const float* __restrict__ tgt,
                                                         float* __restrict__ gsum,
                                                         unsigned int* __restrict__ gcnt,
                                                         int n, int n4) {
  __shared__ float2 hist[BLOCK * STRIDE_REC];  // 22528 B of the 320KB WGP LDS
  const int tid = threadIdx.x;
  float2* rec = &hist[tid * STRIDE_REC];

#pragma unroll
  for (int b = 0; b < BINS; ++b) rec[b] = make_float2(0.0f, 0.0f);
  // No sync needed: each thread touches only its own records until the reduction.

  const v4f* p4 = (const v4f*)pred;
  const v4f* t4 = (const v4f*)tgt;
  const int stride = gridDim.x * BLOCK;

  for (int i = blockIdx.x * BLOCK + tid; i < n4; i += stride) {
    // One-pass 256MB stream: non-temporal so we don't evict useful L2 lines,
    // plus an explicit next-stride prefetch (global_prefetch_b8, speculative).
    __builtin_prefetch(&p4[i + stride], 0, 0);
    __builtin_prefetch(&t4[i + stride], 0, 0);
    v4f p = __builtin_nontemporal_load(&p4[i]);
    v4f t = __builtin_nontemporal_load(&t4[i]);
#pragma unroll
    for (int k = 0; k < 4; ++k) {
      float d = fabsf(p[k] - t[k]);
      float g = sqrtf(d);                    // diff^P, P=0.5, MAX_=1
      int b = (int)(g * 10.0f);              // truncation matches astype(int32)
      b = b > (BINS - 1) ? (BINS - 1) : b;   // last bin absorbs the rest
      float s = d * g;                       // diff^1.5
      float2 v = rec[b];                     // ds_load_b64 (private -> no conflicts)
      v.x += 1.0f;
      v.y += s;
      rec[b] = v;                            // ds_store_b64
    }
  }

  // Scalar tail (N % 4 elements), one global thread each.
  int rem_start = n4 * 4;
  int gid = blockIdx.x * BLOCK + tid;
  if (gid < n - rem_start) {
    float d = fabsf(pred[rem_start + gid] - tgt[rem_start + gid]);
    float g = sqrtf(d);
    int b = (int)(g * 10.0f);
    b = b > (BINS - 1) ? (BINS - 1) : b;
    float2 v = rec[b];
    v.x += 1.0f;
    v.y += d * g;
    rec[b] = v;
  }

  __syncthreads();

  // Tree-fold the 256 private copies (log2(256) = 8 steps).
#pragma unroll 1
  for (int s = BLOCK / 2; s > 0; s >>= 1) {
    if (tid < s) {
      float2* a = &hist[tid * STRIDE_REC];
      float2* c = &hist[(tid + s) * STRIDE_REC];
#pragma unroll
      for (int b = 0; b < BINS; ++b) {
        float2 x = a[b];
        float2 y = c[b];
        x.x += y.x;
        x.y += y.y;
        a[b] = x;
      }
    }
    __syncthreads();
  }

  if (tid < BINS) {
    float2 v = hist[tid];  // thread-0's records
    // Per-block count <= 64*256 + tail: exact in f32; commit as u32 for global exactness.
    atomicAdd(&gcnt[tid], (unsigned int)(v.x + 0.5f));
    atomicAdd(&gsum[tid], v.y);
  }
}

__global__ __launch_bounds__(32) void ghm_final_kernel(const float* __restrict__ gsum,
                                                       const unsigned int* __restrict__ gcnt,
                                                       float* __restrict__ out) {
  if (threadIdx.x == 0) {
    float acc = 0.0f;
    int nbins = 0;
    for (int b = 0; b < BINS; ++b) {
      unsigned int c = gcnt[b];
      if (c != 0u) {
        nbins++;
        acc += gsum[b] / (float)c;
      }
    }
    out[0] = (nbins > 0) ? acc / (float)nbins : 0.0f;
  }
}

extern "C" void kernel_launch(void* const* d_in, const int* in_sizes, int n_in,
                              void* d_out, int out_size, void* d_ws, size_t ws_size,
                              hipStream_t stream) {
  const float* pred = (const float*)d_in[0];
  const float* tgt = (const float*)d_in[1];
  float* out = (float*)d_out;

  float* gsum = (float*)d_ws;                              // 10 floats
  unsigned int* gcnt = (unsigned int*)((char*)d_ws + 64);  // 10 u32, padded apart

  int n = in_sizes[0];
  int n4 = n / 4;

  int grid = 2048;  // 16 float4-iters/thread at N=33.5M; plenty of waves per WGP
  int max_grid = (n4 + BLOCK - 1) / BLOCK;
  if (max_grid < 1) max_grid = 1;
  if (grid > max_grid) grid = max_grid;

  ghm_init_kernel<<<1, BLOCK, 0, stream>>>(gsum, gcnt, out);
  ghm_hist_kernel<<<grid, BLOCK, 0, stream>>>(pred, tgt, gsum, gcnt, n, n4);
  ghm_final_kernel<<<1, 32, 0, stream>>>(gsum, gcnt, out);
}